// JumpKnowGCN_68899865362570
// MI455X (gfx1250) — compile-verified
//
#include <hip/hip_runtime.h>
#include <hip/hip_bf16.h>
#include <stdint.h>

#define N_NODES   100000
#define N_EDGES   1600000
#define IN_DIM    128
#define HID_DIM   64
#define N_CLASSES 40
#define N_HID     3

typedef float v2f __attribute__((ext_vector_type(2)));
typedef float v8f __attribute__((ext_vector_type(8)));

// ---------------- degree / normalization ----------------

__global__ void deg_init_kernel(float* __restrict__ deg) {
    int i = blockIdx.x * 256 + threadIdx.x;
    if (i < N_NODES) deg[i] = 1.0f;           // self-loop contributes 1
}

__global__ void deg_count_kernel(const int* __restrict__ dst, float* __restrict__ deg) {
    int e = blockIdx.x * 256 + threadIdx.x;
    if (e < N_EDGES) atomicAdd(&deg[dst[e]], 1.0f);
}

__global__ void deg_rsqrt_kernel(const float* __restrict__ deg, float* __restrict__ dinv) {
    int i = blockIdx.x * 256 + threadIdx.x;
    if (i < N_NODES) dinv[i] = rsqrtf(deg[i]);
}

// ---------------- WMMA FP32 GEMM: C[N x DOUT] = A[N x K] @ W[K x DOUT] (+bias) ----------------
// One wave32 computes one 16x16 tile with V_WMMA_F32_16X16X4_F32.
// ISA 16x4 f32 A layout: lanes 0-15 -> M=lane,    VGPR0=K+0, VGPR1=K+1;
//                        lanes 16-31 -> M=lane-16, VGPR0=K+2, VGPR1=K+3.
// B (4x16) mirrors: VGPR0 = row K+2*half, VGPR1 = row K+2*half+1, col = lane&15.
// C/D: VGPR j -> row m0 + j + 8*half, col n0 + (lane&15).
//
// Out-of-range columns (head GEMM, DOUT=40): D column n depends only on B column n,
// so we load a CLAMPED in-bounds column unconditionally (no divergent loads) and
// simply never store the dead columns.

template <int K, int DOUT>
__global__ void gemm_wmma_f32_kernel(const float* __restrict__ A,
                                     const float* __restrict__ W,
                                     const float* __restrict__ bias,
                                     float* __restrict__ C)
{
    const int lane = threadIdx.x & 31;
    const int half = lane >> 4;       // 0: rows K+0/K+1   1: rows K+2/K+3
    const int l    = lane & 15;
    const int m0   = blockIdx.x * 16;
    const int n0   = blockIdx.y * 16;
    const int col  = n0 + l;
    const int ccol = (col < DOUT) ? col : (DOUT - 1);   // clamp: dead cols never stored

    const float* __restrict__ arow = A + (size_t)(m0 + l) * K + 2 * half;
    const float* __restrict__ wcol = W + 2 * half * DOUT + ccol;

    __builtin_prefetch(wcol, 0, 1);   // global_prefetch_b8 (non-speculative RT)

    v8f c = {};
#pragma unroll
    for (int k = 0; k < K; k += 4) {
        v2f a, b;
        a.x = arow[k];                 // -> global_load_b64, immediate offset
        a.y = arow[k + 1];
        b.x = wcol[k * DOUT];          // -> global_load_b32, immediate offset
        b.y = wcol[(k + 1) * DOUT];
        // 8 args: (neg_a, A, neg_b, B, c_mod, C, reuse_a, reuse_b)
        c = __builtin_amdgcn_wmma_f32_16x16x4_f32(false, a, false, b,
                                                  (short)0, c, false, false);
    }

    if (col < DOUT) {
        const float bb = bias ? bias[col] : 0.0f;
        float* __restrict__ outp = C + (size_t)(m0 + half * 8) * DOUT + col;
#pragma unroll
        for (int j = 0; j < 8; ++j)
            outp[j * DOUT] = c[j] + bb;
    }
}

// ---------------- aggregation (float4-vectorized: b128 traffic) ----------------

// acc[i, f..f+3] = bias[f..f+3] + hw[i, f..f+3] * dinv[i]^2   (self-loop + bias)
__global__ void acc_init_kernel(const float4* __restrict__ hw4,
                                const float* __restrict__ dinv,
                                const float* __restrict__ bias,
                                float4* __restrict__ acc4)
{
    int idx = blockIdx.x * 256 + threadIdx.x;          // over N_NODES * 16
    if (idx >= N_NODES * (HID_DIM / 4)) return;
    int i  = idx >> 4;
    int f  = (idx & 15) * 4;
    float di = dinv[i];
    float w  = di * di;
    float4 h = hw4[idx];
    float4 r;
    r.x = bias[f + 0] + h.x * w;
    r.y = bias[f + 1] + h.y * w;
    r.z = bias[f + 2] + h.z * w;
    r.w = bias[f + 3] + h.w * w;
    acc4[idx] = r;
}

// acc[dst, f..f+3] += hw[src, f..f+3] * dinv[src] * dinv[dst]; one thread per (edge, 4 features)
__global__ void scatter_kernel(const float* __restrict__ hw,
                               const int* __restrict__ src,
                               const int* __restrict__ dst,
                               const float* __restrict__ dinv,
                               float* __restrict__ acc)
{
    long long idx = (long long)blockIdx.x * 256 + threadIdx.x;   // over E * 16
    if (idx >= (long long)N_EDGES * (HID_DIM / 4)) return;
    int e = (int)(idx >> 4);
    int f = (int)(idx & 15) * 4;
    int s = src[e];
    int d = dst[e];
    float nrm = dinv[s] * dinv[d];
    float4 h = *(const float4*)(hw + (size_t)s * HID_DIM + f);   // global_load_b128
    float* ap = acc + (size_t)d * HID_DIM + f;
    atomicAdd(ap + 0, h.x * nrm);
    atomicAdd(ap + 1, h.y * nrm);
    atomicAdd(ap + 2, h.z * nrm);
    atomicAdd(ap + 3, h.w * nrm);
}

// h = relu(acc); jk = first ? h : max(jk, h)
__global__ void relu_jk_kernel(const float4* __restrict__ acc4,
                               float4* __restrict__ h4,
                               float4* __restrict__ jk4,
                               int first)
{
    int idx = blockIdx.x * 256 + threadIdx.x;          // over N_NODES * 16
    if (idx >= N_NODES * (HID_DIM / 4)) return;
    float4 v = acc4[idx];
    v.x = fmaxf(v.x, 0.0f);
    v.y = fmaxf(v.y, 0.0f);
    v.z = fmaxf(v.z, 0.0f);
    v.w = fmaxf(v.w, 0.0f);
    h4[idx] = v;
    if (first) {
        jk4[idx] = v;
    } else {
        float4 m = jk4[idx];
        m.x = fmaxf(m.x, v.x);
        m.y = fmaxf(m.y, v.y);
        m.z = fmaxf(m.z, v.z);
        m.w = fmaxf(m.w, v.w);
        jk4[idx] = m;
    }
}

// ---------------- launch ----------------

extern "C" void kernel_launch(void* const* d_in, const int* in_sizes, int n_in,
                              void* d_out, int out_size, void* d_ws, size_t ws_size,
                              hipStream_t stream)
{
    const float* x  = (const float*)d_in[0];   // [N, 128]
    const int*   ei = (const int*)  d_in[1];   // [2, E]
    const float* W0 = (const float*)d_in[2];   // [128, 64]
    const float* b0 = (const float*)d_in[3];   // [64]
    const float* Wh = (const float*)d_in[4];   // [3, 64, 64]
    const float* bh = (const float*)d_in[5];   // [3, 64]
    const float* Wp = (const float*)d_in[6];   // [64, 40]
    const float* bp = (const float*)d_in[7];   // [40]
    const int* src = ei;
    const int* dst = ei + N_EDGES;
    float* out = (float*)d_out;

    // carve workspace (256B-aligned regions)
    char*  ws  = (char*)d_ws;
    size_t off = 0;
    auto carve = [&](size_t bytes) -> float* {
        float* p = (float*)(ws + off);
        off += (bytes + 255) & ~(size_t)255;
        return p;
    };
    float* deg  = carve((size_t)N_NODES * sizeof(float));
    float* dinv = carve((size_t)N_NODES * sizeof(float));
    float* hw   = carve((size_t)N_NODES * HID_DIM * sizeof(float));
    float* acc  = carve((size_t)N_NODES * HID_DIM * sizeof(float));
    float* h    = carve((size_t)N_NODES * HID_DIM * sizeof(float));
    float* jk   = carve((size_t)N_NODES * HID_DIM * sizeof(float));

    const dim3 blk(256);
    const unsigned gN   = (N_NODES + 255) / 256;
    const unsigned gE   = (N_EDGES + 255) / 256;
    const unsigned gNF4 = (unsigned)(((long long)N_NODES * (HID_DIM / 4) + 255) / 256);
    const unsigned gEF4 = (unsigned)(((long long)N_EDGES * (HID_DIM / 4) + 255) / 256);

    // degree / normalization
    deg_init_kernel <<<gN, blk, 0, stream>>>(deg);
    deg_count_kernel<<<gE, blk, 0, stream>>>(dst, deg);
    deg_rsqrt_kernel<<<gN, blk, 0, stream>>>(deg, dinv);

    const dim3 gemm_grid(N_NODES / 16, HID_DIM / 16);   // 6250 x 4
    const dim3 gemm_blk(32);                            // one wave32 per tile

    // layer 0: x @ W0
    gemm_wmma_f32_kernel<IN_DIM, HID_DIM><<<gemm_grid, gemm_blk, 0, stream>>>(x, W0, nullptr, hw);
    acc_init_kernel<<<gNF4, blk, 0, stream>>>((const float4*)hw, dinv, b0, (float4*)acc);
    scatter_kernel <<<gEF4, blk, 0, stream>>>(hw, src, dst, dinv, acc);
    relu_jk_kernel <<<gNF4, blk, 0, stream>>>((const float4*)acc, (float4*)h, (float4*)jk, 1);

    // hidden layers
    for (int i = 0; i < N_HID; ++i) {
        gemm_wmma_f32_kernel<HID_DIM, HID_DIM><<<gemm_grid, gemm_blk, 0, stream>>>(
            h, Wh + (size_t)i * HID_DIM * HID_DIM, nullptr, hw);
        acc_init_kernel<<<gNF4, blk, 0, stream>>>((const float4*)hw, dinv,
                                                  bh + (size_t)i * HID_DIM, (float4*)acc);
        scatter_kernel <<<gEF4, blk, 0, stream>>>(hw, src, dst, dinv, acc);
        relu_jk_kernel <<<gNF4, blk, 0, stream>>>((const float4*)acc, (float4*)h, (float4*)jk, 0);
    }

    // head: jk @ Wp + bp -> d_out [N, 40]; 3 column tiles, last one column-masked on store
    const dim3 head_grid(N_NODES / 16, (N_CLASSES + 15) / 16);
    gemm_wmma_f32_kernel<HID_DIM, N_CLASSES><<<head_grid, gemm_blk, 0, stream>>>(jk, Wp, bp, out);
}